// GATLayer_91216515432633
// MI455X (gfx1250) — compile-verified
//
#include <hip/hip_runtime.h>
#include <math.h>

// GAT layer, MI455X (gfx1250, wave32).
// Inputs: q[N,64] f32, W_att[128], W_upd[64,64], W_V[64,64], b_V[64],
//         src1,dst1,src2,dst2 [E] i32
// Outputs (concat): q_new[N,64], OV[N,64]
//
// Uses linearity: OV = (OK1+OK2)@W_V^T + (s1+s2)*b_V ; q_new = (2q + OK1+OK2)@W_upd^T
// so only ONE edge-scatter buffer X = OK1+OK2 is needed, and all GEMMs are
// node-level [N,64]x[64,64] done with V_WMMA_F32_16X16X4_F32 (full f32 precision).

typedef __attribute__((ext_vector_type(2))) float v2f;
typedef __attribute__((ext_vector_type(8))) float v8f;

__device__ __forceinline__ float fatomic_add(float* p, float v) {
    return __hip_atomic_fetch_add(p, v, __ATOMIC_RELAXED, __HIP_MEMORY_SCOPE_AGENT);
}

// ---------------- zero workspace ----------------
__global__ void k_zero(float* __restrict__ p, long long n) {
    long long i = (long long)blockIdx.x * blockDim.x + threadIdx.x;
    long long stride = (long long)gridDim.x * blockDim.x;
    for (; i < n; i += stride) p[i] = 0.0f;
}

// ---------------- per-node attention dot products ----------------
// attS[i] = q[i,:] . W_att[0:64] ; attD[i] = q[i,:] . W_att[64:128]
// One wave (32 lanes) per node, 2 elements per lane, shuffle reduce.
__global__ __launch_bounds__(256) void k_att(const float* __restrict__ q,
                                             const float* __restrict__ Watt,
                                             float* __restrict__ attS,
                                             float* __restrict__ attD, int N) {
    int lane = threadIdx.x & 31;
    int node = (blockIdx.x * blockDim.x + threadIdx.x) >> 5;
    if (node >= N) return;
    const float* row = q + (long long)node * 64;
    float q0 = row[lane], q1 = row[lane + 32];
    float s = q0 * Watt[lane]      + q1 * Watt[lane + 32];
    float d = q0 * Watt[64 + lane] + q1 * Watt[96 + lane];
    #pragma unroll
    for (int off = 16; off > 0; off >>= 1) {
        s += __shfl_xor(s, off, 32);
        d += __shfl_xor(d, off, 32);
    }
    if (lane == 0) { attS[node] = s; attD[node] = d; }
}

// ---------------- per-edge softmax denominator ----------------
__global__ __launch_bounds__(256) void k_denom(const int* __restrict__ src,
                                               const int* __restrict__ dst,
                                               const float* __restrict__ attS,
                                               const float* __restrict__ attD,
                                               float* __restrict__ den, int E) {
    int i = blockIdx.x * blockDim.x + threadIdx.x;
    int stride = gridDim.x * blockDim.x;
    for (int e = i; e < E; e += stride) {
        int s = src[e], d = dst[e];
        float x  = attS[s] + attD[d];
        float sg = 1.0f / (1.0f + expf(-x));
        fatomic_add(den + d, expf(sg));   // ex = exp(sigmoid(x)) in (1, e)
    }
}

// ---------------- weighted scatter: X[dst,:] += a * q[src,:] ----------------
// One wave handles 32 consecutive edges: per-lane edge metadata (coalesced),
// then per edge all 32 lanes move a float2 (row gather + 2 f32 atomics).
__global__ __launch_bounds__(256) void k_scatter(const int* __restrict__ src,
                                                 const int* __restrict__ dst,
                                                 const float* __restrict__ attS,
                                                 const float* __restrict__ attD,
                                                 const float* __restrict__ den,
                                                 const float* __restrict__ q,
                                                 float* __restrict__ X, int E) {
    int lane = threadIdx.x & 31;
    int wave = (blockIdx.x * blockDim.x + threadIdx.x) >> 5;
    int base = wave * 32;
    if (base >= E) return;

    int s = 0, d = 0; float a = 0.0f;
    int e = base + lane;
    if (e < E) {
        s = src[e]; d = dst[e];
        float x  = attS[s] + attD[d];
        float sg = 1.0f / (1.0f + expf(-x));
        float ex = expf(sg);
        a = ex / fmaxf(den[d], 1e-12f);
    }
    const float2* q2 = (const float2*)q;
    int cnt = min(32, E - base);
    for (int i = 0; i < cnt; ++i) {
        float ai = __shfl(a, i, 32);
        int   si = __shfl(s, i, 32);
        int   di = __shfl(d, i, 32);
        float2 v = q2[si * 32 + lane];          // 8B per lane, 256B/row coalesced
        float* xp = X + (long long)di * 64 + lane * 2;
        fatomic_add(xp,     ai * v.x);
        fatomic_add(xp + 1, ai * v.y);
    }
}

// ---------------- fused dual GEMM with WMMA ----------------
// q_new = (2q + X) @ W_upd^T ;  OV = X @ W_V^T + (1{den1>0}+1{den2>0}) * b_V
// Block = 256 thr = 8 waves, covers 32 output rows x 64 cols.
// Wave w: rows r0 = blk*32 + (w>>2)*16, cols c0 = (w&3)*16.
// K=64 via 16 chained v_wmma_f32_16x16x4_f32 per accumulator.
// A frag (16x4): lanes 0-15 supply {A[r0+l, k], A[r0+l, k+1]},
//                lanes 16-31 supply {A[r0+l, k+2], A[r0+l, k+3]}.
// B frag (4x16): B[kk,n] = W[c0+n, k+kk] -> same float2 pattern indexed by n.
__global__ __launch_bounds__(256) void k_gemm(const float* __restrict__ q,
                                              const float* __restrict__ X,
                                              const float* __restrict__ Wupd,
                                              const float* __restrict__ Wv,
                                              const float* __restrict__ bV,
                                              const float* __restrict__ den1,
                                              const float* __restrict__ den2,
                                              float* __restrict__ out, int N) {
    __shared__ float sWu[64 * 64];
    __shared__ float sWv[64 * 64];
    for (int i = threadIdx.x; i < 64 * 64; i += 256) {
        sWu[i] = Wupd[i];
        sWv[i] = Wv[i];
    }
    __syncthreads();

    int lane = threadIdx.x & 31;
    int w    = threadIdx.x >> 5;
    int r0   = blockIdx.x * 32 + (w >> 2) * 16;
    int c0   = (w & 3) * 16;
    int half = lane >> 4;          // 0: K lo pair, 1: K hi pair
    int l    = lane & 15;

    int arow  = r0 + l;
    int arow_c = min(arow, N - 1); // clamp loads for ragged tail (stores guarded)
    int kofs  = half * 2;

    v8f accU = {};                 // (2q + X) @ W_upd^T
    v8f accV = {};                 // X @ W_V^T

    #pragma unroll
    for (int k = 0; k < 64; k += 4) {
        float2 xv = *(const float2*)(X + (long long)arow_c * 64 + k + kofs);
        float2 qv = *(const float2*)(q + (long long)arow_c * 64 + k + kofs);
        v2f a1; a1.x = 2.0f * qv.x + xv.x; a1.y = 2.0f * qv.y + xv.y;
        v2f a2; a2.x = xv.x;               a2.y = xv.y;
        v2f bu = *(const v2f*)&sWu[(c0 + l) * 64 + k + kofs];
        v2f bv = *(const v2f*)&sWv[(c0 + l) * 64 + k + kofs];
        accU = __builtin_amdgcn_wmma_f32_16x16x4_f32(false, a1, false, bu,
                                                     (short)0, accU, false, false);
        accV = __builtin_amdgcn_wmma_f32_16x16x4_f32(false, a2, false, bv,
                                                     (short)0, accV, false, false);
    }

    // D layout: VGPR r -> row r0 + r + half*8, col c0 + (lane&15)
    int col = c0 + l;
    float bvc = bV[col];
    #pragma unroll
    for (int r = 0; r < 8; ++r) {
        int row = r0 + r + half * 8;
        if (row < N) {
            float sI = ((den1[row] > 0.0f) ? 1.0f : 0.0f)
                     + ((den2[row] > 0.0f) ? 1.0f : 0.0f);
            out[(long long)row * 64 + col] = accU[r];
            out[(long long)N * 64 + (long long)row * 64 + col] = accV[r] + sI * bvc;
        }
    }
}

extern "C" void kernel_launch(void* const* d_in, const int* in_sizes, int n_in,
                              void* d_out, int out_size, void* d_ws, size_t ws_size,
                              hipStream_t stream) {
    const float* q    = (const float*)d_in[0];
    const float* Watt = (const float*)d_in[1];
    const float* Wupd = (const float*)d_in[2];
    const float* Wv   = (const float*)d_in[3];
    const float* bV   = (const float*)d_in[4];
    const int*   src1 = (const int*)d_in[5];
    const int*   dst1 = (const int*)d_in[6];
    const int*   src2 = (const int*)d_in[7];
    const int*   dst2 = (const int*)d_in[8];

    const int N = in_sizes[0] / 64;
    const int E = in_sizes[5];

    // Workspace layout (floats): X[N*64] | attS[N] | attD[N] | den1[N] | den2[N]
    float* X    = (float*)d_ws;
    float* attS = X + (long long)N * 64;
    float* attD = attS + N;
    float* den1 = attD + N;
    float* den2 = den1 + N;

    float* out = (float*)d_out;

    long long zeroN = (long long)N * 64 + 4LL * N;
    k_zero<<<2048, 256, 0, stream>>>((float*)d_ws, zeroN);

    int attBlocks = (N + 7) / 8;                 // 8 waves/block, wave per node
    k_att<<<attBlocks, 256, 0, stream>>>(q, Watt, attS, attD, N);

    int edgeBlocks = (E + 255) / 256;
    k_denom<<<edgeBlocks, 256, 0, stream>>>(src1, dst1, attS, attD, den1, E);
    k_denom<<<edgeBlocks, 256, 0, stream>>>(src2, dst2, attS, attD, den2, E);

    int waves = (E + 31) / 32;
    int scatterBlocks = (waves + 7) / 8;         // 8 waves/block
    k_scatter<<<scatterBlocks, 256, 0, stream>>>(src1, dst1, attS, attD, den1, q, X, E);
    k_scatter<<<scatterBlocks, 256, 0, stream>>>(src2, dst2, attS, attD, den2, q, X, E);

    int gemmBlocks = (N + 31) / 32;              // 32 rows per block
    k_gemm<<<gemmBlocks, 256, 0, stream>>>(q, X, Wupd, Wv, bV, den1, den2, out, N);
}